// OnlineTripletLoss_55929064128529
// MI455X (gfx1250) — compile-verified
//
#include <hip/hip_runtime.h>
#include <hip/hip_bf16.h>

typedef __attribute__((ext_vector_type(2))) float v2f;
typedef __attribute__((ext_vector_type(8))) float v8f;

#define NN 512
#define DD 512
#define MARGIN_F 1.0f

// ---------------------------------------------------------------------------
// Kernel 0: sq[i] = ||E[i,:]||^2 ; one wave (32 lanes) per row
// ---------------------------------------------------------------------------
__global__ __launch_bounds__(32) void rownorm_kernel(const float* __restrict__ E,
                                                     float* __restrict__ sq) {
    const int row  = blockIdx.x;
    const int lane = threadIdx.x;
    const float* r = E + row * DD;
    float s = 0.0f;
    #pragma unroll 4
    for (int j = lane; j < DD; j += 32) {
        float v = r[j];
        s = fmaf(v, v, s);
    }
    #pragma unroll
    for (int off = 16; off > 0; off >>= 1)
        s += __shfl_xor(s, off, 32);
    if (lane == 0) sq[row] = s;
}

// ---------------------------------------------------------------------------
// Kernel 1: dist[i,j] = sq[i] + sq[j] - 2 * (E E^T)[i,j]
// One wave per 16x16 tile, fp32 WMMA 16x16x4 along K.
//
// A-matrix (16x4 f32) lane layout per ISA:
//   lanes 0-15  : VGPR0 = A[m=lane, k],   VGPR1 = A[m=lane, k+1]
//   lanes 16-31 : VGPR0 = A[m=lane-16, k+2], VGPR1 = A[m=lane-16, k+3]
// B = E^T restricted to rows J..J+15, same per-lane pattern with row base J.
// C/D (16x16 f32): acc[v] of lane l -> m = v + (l<16?0:8), n = l&15.
// ---------------------------------------------------------------------------
__global__ __launch_bounds__(32) void gram_dist_kernel(const float* __restrict__ E,
                                                       const float* __restrict__ sq,
                                                       float* __restrict__ dist) {
    const int ti   = blockIdx.x * 16;   // row-tile base (M / anchor i)
    const int tj   = blockIdx.y * 16;   // col-tile base (N / index j)
    const int lane = threadIdx.x;       // 0..31, full wave (EXEC all ones)
    const int r    = lane & 15;
    const int kofs = (lane >> 4) * 2;   // 0 for lanes 0-15, 2 for lanes 16-31

    const float* arow = E + (ti + r) * DD + kofs;
    const float* brow = E + (tj + r) * DD + kofs;

    v8f acc = {};
    #pragma unroll 8
    for (int k = 0; k < DD; k += 4) {
        v2f a = *(const v2f*)(arow + k);   // 8-byte contiguous per lane
        v2f b = *(const v2f*)(brow + k);
        acc = __builtin_amdgcn_wmma_f32_16x16x4_f32(
            /*neg_a=*/false, a, /*neg_b=*/false, b,
            /*c_mod=*/(short)0, acc, /*reuse_a=*/false, /*reuse_b=*/false);
    }

    const int n     = lane & 15;
    const int mbase = (lane >> 4) * 8;
    const float sqn = sq[tj + n];
    #pragma unroll
    for (int v = 0; v < 8; ++v) {
        const int m = mbase + v;
        dist[(ti + m) * NN + (tj + n)] = sq[ti + m] + sqn - 2.0f * acc[v];
    }
}

// ---------------------------------------------------------------------------
// Kernel 2: per-anchor masked triplet sum. One block (256 thr) per anchor.
// Only row a of dist is needed:  sum_{p in P(a)} sum_{n in N(a)}
//     relu(dist[a,p] - dist[a,n] + margin)
// Positives compacted serially by thread 0 (deterministic order, no atomics).
// neg count = 511 - pos count (exact), so count_a = pos * (511 - pos).
// ---------------------------------------------------------------------------
__global__ __launch_bounds__(256) void triplet_kernel(const int*   __restrict__ labels,
                                                      const float* __restrict__ dist,
                                                      float* __restrict__ psum,
                                                      float* __restrict__ pcnt) {
    __shared__ int   lab[NN];
    __shared__ float posd[NN];
    __shared__ int   s_poscnt;
    __shared__ float red[256];

    const int a = blockIdx.x;
    const int t = threadIdx.x;

    for (int j = t; j < NN; j += 256) lab[j] = labels[j];
    __syncthreads();

    const int la = lab[a];
    const float* drow = dist + a * NN;

    if (t == 0) {
        int pc = 0;
        for (int j = 0; j < NN; ++j)
            if (j != a && lab[j] == la) posd[pc++] = drow[j];
        s_poscnt = pc;
    }
    __syncthreads();

    const int pc = s_poscnt;
    float s = 0.0f;
    for (int j = t; j < NN; j += 256) {
        if (lab[j] != la) {
            const float base = MARGIN_F - drow[j];   // margin - d_an
            for (int q = 0; q < pc; ++q) {
                const float term = posd[q] + base;
                s += term > 0.0f ? term : 0.0f;
            }
        }
    }

    red[t] = s;
    __syncthreads();
    #pragma unroll
    for (int off = 128; off > 0; off >>= 1) {
        if (t < off) red[t] += red[t + off];
        __syncthreads();
    }
    if (t == 0) {
        psum[a] = red[0];
        pcnt[a] = (float)(pc * (NN - 1 - pc));
    }
}

// ---------------------------------------------------------------------------
// Kernel 3: final deterministic reduction of 512 per-anchor partials.
// ---------------------------------------------------------------------------
__global__ __launch_bounds__(256) void finalize_kernel(const float* __restrict__ psum,
                                                       const float* __restrict__ pcnt,
                                                       float* __restrict__ out) {
    __shared__ float rs[256];
    __shared__ float rc[256];
    const int t = threadIdx.x;
    rs[t] = psum[t] + psum[t + 256];
    rc[t] = pcnt[t] + pcnt[t + 256];
    __syncthreads();
    #pragma unroll
    for (int off = 128; off > 0; off >>= 1) {
        if (t < off) { rs[t] += rs[t + off]; rc[t] += rc[t + off]; }
        __syncthreads();
    }
    if (t == 0) {
        float cnt = rc[0] < 1.0f ? 1.0f : rc[0];
        out[0] = rs[0] / cnt;
    }
}

// ---------------------------------------------------------------------------
extern "C" void kernel_launch(void* const* d_in, const int* in_sizes, int n_in,
                              void* d_out, int out_size, void* d_ws, size_t ws_size,
                              hipStream_t stream) {
    const float* E      = (const float*)d_in[0];   // [512,512] fp32
    const int*   labels = (const int*)d_in[1];     // [512] int
    float*       out    = (float*)d_out;           // scalar

    char*  ws   = (char*)d_ws;
    float* sq   = (float*)ws;                               // 512 f
    float* dist = (float*)(ws + 4096);                      // 512*512 f = 1 MB
    float* psum = (float*)(ws + 4096 + NN * NN * 4);        // 512 f
    float* pcnt = psum + NN;                                // 512 f

    rownorm_kernel  <<<NN, 32, 0, stream>>>(E, sq);
    gram_dist_kernel<<<dim3(NN / 16, NN / 16), 32, 0, stream>>>(E, sq, dist);
    triplet_kernel  <<<NN, 256, 0, stream>>>(labels, dist, psum, pcnt);
    finalize_kernel <<<1, 256, 0, stream>>>(psum, pcnt, out);
}